// VoxelOutBlock_89498528514659
// MI455X (gfx1250) — compile-verified
//
#include <hip/hip_runtime.h>
#include <hip/hip_bf16.h>
#include <math.h>

// Problem structure (fixed by reference): B=32, IN_PLANES=128, PLANES=32.
// n_sel passed at launch (from in_sizes of voxel_indices).

typedef __attribute__((ext_vector_type(2))) float v2f;
typedef __attribute__((ext_vector_type(8))) float v8f;

#define BATCH      32
#define IN_PLANES  128
#define PLANES     32
#define A_STRIDE   132   // 128 + 4 pad: lane m hits bank 4m (+k) -> conflict-free
#define W_STRIDE   132   // transposed W0: Wt[o][i], contiguous K per output col
#define H_STRIDE   33

__global__ __launch_bounds__(128) void voxel_mlp_kernel(
    const float* __restrict__ x,     // [32, n_sel, 128]
    const int*   __restrict__ vidx,  // [n_sel]
    const float* __restrict__ w0,    // [n_vox, 128, 32]
    const float* __restrict__ b0,    // [n_vox, 32]
    const float* __restrict__ w1,    // [n_vox, 32]
    const float* __restrict__ b1,    // [n_vox]
    float* __restrict__ out,         // [32, n_sel]
    int n_sel)
{
    __shared__ float As[BATCH * A_STRIDE];   // x tile  [b][i]
    __shared__ float Wt[PLANES * W_STRIDE];  // w0 tile transposed [o][i]
    __shared__ float Hs[BATCH * H_STRIDE];   // hidden  [b][o]
    __shared__ float W1s[PLANES];
    __shared__ float B1s;

    const int n   = blockIdx.x;       // selected voxel
    const int tid = threadIdx.x;      // 0..127
    const int v   = vidx[n];

    // ---- cooperative staging: x tile (32x128) and w0 tile (128x32) ----
    const size_t x_bstride = (size_t)n_sel * IN_PLANES;
    const float* xbase  = x  + (size_t)n * IN_PLANES;
    const float* w0base = w0 + (size_t)v * (IN_PLANES * PLANES);

    #pragma unroll
    for (int it = 0; it < 8; ++it) {
        // A: 1024 float4 slots; 32 float4 per batch row
        int l   = it * 128 + tid;
        int row = l >> 5;                 // batch index 0..31
        int col = (l & 31) << 2;          // 0,4,...,124
        float4 fa = *(const float4*)(xbase + (size_t)row * x_bstride + col);
        *(float4*)(&As[row * A_STRIDE + col]) = fa;

        // W0: 1024 float4 slots; 8 float4 per K-row of 32 outputs. Transpose on store.
        int r  = l >> 3;                  // k index 0..127
        int nc = (l & 7) << 2;            // output col 0,4,...,28
        float4 fw = *(const float4*)(w0base + r * PLANES + nc);
        Wt[(nc + 0) * W_STRIDE + r] = fw.x;
        Wt[(nc + 1) * W_STRIDE + r] = fw.y;
        Wt[(nc + 2) * W_STRIDE + r] = fw.z;
        Wt[(nc + 3) * W_STRIDE + r] = fw.w;
    }
    if (tid < PLANES) W1s[tid] = w1[(size_t)v * PLANES + tid];
    if (tid == 0)     B1s = b1[v];
    __syncthreads();

    // ---- WMMA: each wave32 computes one 16x16 tile of the 32x32 hidden ----
    const int wave   = tid >> 5;          // 0..3
    const int lane   = tid & 31;
    const int mi     = wave >> 1;         // M tile 0/1
    const int ni     = wave & 1;          // N tile 0/1
    const int hl     = lane & 15;         // row (A) / col (B) within tile
    const int laneHi = lane >> 4;         // 0: K+0/M+0 half, 1: K+2/M+8 half

    const float* Arow = &As[(mi * 16 + hl) * A_STRIDE + 2 * laneHi];
    const float* Brow = &Wt[(ni * 16 + hl) * W_STRIDE + 2 * laneHi];

    v8f acc = {};
    #pragma unroll 8
    for (int kk = 0; kk < IN_PLANES; kk += 4) {
        // A 16x4 f32 fragment: v0=A[m][k], v1=A[m][k+1]; hi lanes K+2
        v2f a = *(const v2f*)(Arow + kk);
        // B 4x16 f32 fragment: v0=W[k][nc], v1=W[k+1][nc]; hi lanes K+2
        v2f b = *(const v2f*)(Brow + kk);
        acc = __builtin_amdgcn_wmma_f32_16x16x4_f32(
            /*neg_a=*/false, a, /*neg_b=*/false, b,
            /*c_mod=*/(short)0, acc, /*reuse_a=*/false, /*reuse_b=*/false);
    }

    // ---- scale 1/128 + bias + exact GELU, spill to LDS ----
    const int ncol = ni * 16 + hl;
    const float bias = b0[(size_t)v * PLANES + ncol];
    #pragma unroll
    for (int r = 0; r < 8; ++r) {
        float h = acc[r] * (1.0f / (float)IN_PLANES) + bias;
        h = 0.5f * h * (1.0f + erff(h * 0.70710678118654752f));
        Hs[(mi * 16 + r + 8 * laneHi) * H_STRIDE + ncol] = h;
    }
    __syncthreads();

    // ---- layer 1: out[b] = (h[b,:] . w1) / 32 + b1 ----
    if (tid < BATCH) {
        float s = 0.0f;
        #pragma unroll
        for (int p = 0; p < PLANES; ++p)
            s += Hs[tid * H_STRIDE + p] * W1s[p];
        out[(size_t)tid * n_sel + n] = s * (1.0f / (float)PLANES) + B1s;
    }
}

extern "C" void kernel_launch(void* const* d_in, const int* in_sizes, int n_in,
                              void* d_out, int out_size, void* d_ws, size_t ws_size,
                              hipStream_t stream) {
    const float* x    = (const float*)d_in[0];
    const int*   vidx = (const int*)  d_in[1];
    const float* w0   = (const float*)d_in[2];
    const float* b0   = (const float*)d_in[3];
    const float* w1   = (const float*)d_in[4];
    const float* b1   = (const float*)d_in[5];
    float* out = (float*)d_out;

    const int n_sel = in_sizes[1];   // number of selected voxels (20000)

    voxel_mlp_kernel<<<n_sel, 128, 0, stream>>>(x, vidx, w0, b0, w1, b1, out, n_sel);
}